// PoolTo_layer_88149908783175
// MI455X (gfx1250) — compile-verified
//
#include <hip/hip_runtime.h>

typedef __attribute__((ext_vector_type(2))) float v2f;
typedef __attribute__((ext_vector_type(8))) float v8f;

#define B_ 8
#define N_ 4096
#define C_ 256
#define P_ 1024
#define K_ 16

// ---------------- Kernel 1: farthest point sampling + gather vertices ------
// One block per batch. Vertices cached in LDS (48KB). Per-thread running
// min-distance kept in registers (16 points/thread, strided). Block argmax
// via wave32 shuffle + LDS tree, ties broken toward lower index (jnp.argmax).
__global__ __launch_bounds__(256) void fps_kernel(const float* __restrict__ verts,
                                                  float* __restrict__ out_v,
                                                  int* __restrict__ sampled)
{
    __shared__ float sx[N_], sy[N_], sz[N_];
    __shared__ float rd[8];
    __shared__ int   ri[8];
    __shared__ int   s_far;

    const int b   = blockIdx.x;
    const int tid = threadIdx.x;
    const float* vb = verts + (size_t)b * N_ * 3;

    for (int i = tid; i < N_; i += 256) {
        sx[i] = vb[i * 3 + 0];
        sy[i] = vb[i * 3 + 1];
        sz[i] = vb[i * 3 + 2];
    }
    __syncthreads();

    float dist[16];
#pragma unroll
    for (int j = 0; j < 16; ++j) dist[j] = 1e10f;

    int far = 0;
    for (int s = 0; s < P_; ++s) {
        if (tid == 0) {
            sampled[b * P_ + s] = far;
            out_v[(b * P_ + s) * 3 + 0] = sx[far];
            out_v[(b * P_ + s) * 3 + 1] = sy[far];
            out_v[(b * P_ + s) * 3 + 2] = sz[far];
        }
        const float cx = sx[far], cy = sy[far], cz = sz[far];

        float bestd = -1.0f;
        int   besti = 0;
#pragma unroll
        for (int j = 0; j < 16; ++j) {
            const int i = tid + j * 256;
            const float dx = sx[i] - cx, dy = sy[i] - cy, dz = sz[i] - cz;
            const float d  = dx * dx + dy * dy + dz * dz;
            const float nd = fminf(dist[j], d);
            dist[j] = nd;
            if (nd > bestd || (nd == bestd && i < besti)) { bestd = nd; besti = i; }
        }
        // wave32 maxloc reduction
#pragma unroll
        for (int off = 16; off > 0; off >>= 1) {
            const float od = __shfl_down(bestd, off, 32);
            const int   oi = __shfl_down(besti, off, 32);
            if (od > bestd || (od == bestd && oi < besti)) { bestd = od; besti = oi; }
        }
        const int wid = tid >> 5;
        if ((tid & 31) == 0) { rd[wid] = bestd; ri[wid] = besti; }
        __syncthreads();
        if (tid == 0) {
            float md = rd[0]; int mi = ri[0];
#pragma unroll
            for (int w = 1; w < 8; ++w)
                if (rd[w] > md || (rd[w] == md && ri[w] < mi)) { md = rd[w]; mi = ri[w]; }
            s_far = mi;
        }
        __syncthreads();
        far = s_far;
        // rd/ri are read before the barrier above; next s_far write happens only after the next iteration's
        // first __syncthreads, so no extra barrier needed here
    }
}

// ---------------- Kernel 2: WMMA distance tiles + running top-16 -----------
// One wave (32 threads) per 16 sampled rows. dist = |vi|^2 + |vj|^2 - 2 vi.vj
// computed as a single V_WMMA_F32_16X16X4_F32:
//   A row  = ( x_i,  y_i,  z_i, 1 )               (16x4)
//   B col  = (-2x_j,-2y_j,-2z_j, |vj|^2 )         (4x16)
//   C      = |vi|^2 broadcast along the row       (16x16)
// Lanes 0-15 each own one row and keep a sorted top-16 (dist,idx) list in
// registers, excluding the self index.
__global__ __launch_bounds__(32) void knn_kernel(const float* __restrict__ verts,
                                                 const int* __restrict__ sampled,
                                                 int* __restrict__ knn)
{
    __shared__ float tileD[16 * 17];
    __shared__ float s_sq[16];

    const int blk  = blockIdx.x;       // b * 64 + t
    const int b    = blk >> 6;
    const int t    = blk & 63;
    const int lane = threadIdx.x;
    const int m    = lane & 15;
    const int half = lane >> 4;        // K half: 0 -> K=0,1 ; 1 -> K=2,3

    const float* vb = verts + (size_t)b * N_ * 3;

    const int   row = sampled[b * P_ + t * 16 + m];
    const float ax = vb[row * 3 + 0], ay = vb[row * 3 + 1], az = vb[row * 3 + 2];
    if (half == 0) s_sq[m] = ax * ax + ay * ay + az * az;
    __syncthreads();

    // A operand (16x4 f32): lanes 0-15 hold K=0,1 ; lanes 16-31 hold K=2,3
    v2f a;
    if (half == 0) { a.x = ax; a.y = ay;   }
    else           { a.x = az; a.y = 1.0f; }

    // C operand (16x16 f32): VGPR r -> M=r (lanes 0-15), M=r+8 (lanes 16-31)
    v8f cacc;
#pragma unroll
    for (int r = 0; r < 8; ++r) cacc[r] = s_sq[half ? (r + 8) : r];

    const int myrow = row;
    float bd[K_]; int bi[K_];
#pragma unroll
    for (int q = 0; q < K_; ++q) { bd[q] = 3.4e38f; bi[q] = 0; }

    for (int ct = 0; ct < N_ / 16; ++ct) {
        const int cb = ct * 16;
        const int j  = cb + m;
        const float wx = vb[j * 3 + 0], wy = vb[j * 3 + 1], wz = vb[j * 3 + 2];
        const float sqj = wx * wx + wy * wy + wz * wz;

        // B operand (4x16 f32): lanes 0-15 hold K=0,1 ; lanes 16-31 K=2,3
        v2f bv;
        if (half == 0) { bv.x = -2.0f * wx; bv.y = -2.0f * wy; }
        else           { bv.x = -2.0f * wz; bv.y = sqj;        }

        v8f d = __builtin_amdgcn_wmma_f32_16x16x4_f32(
            /*neg_a=*/false, a, /*neg_b=*/false, bv,
            /*c_mod=*/(short)0, cacc, /*reuse_a=*/false, /*reuse_b=*/false);

#pragma unroll
        for (int r = 0; r < 8; ++r)
            tileD[(half ? (r + 8) : r) * 17 + m] = d[r];
        __syncthreads();   // single-wave workgroup: barrier lowers to S_NOP,
                           // but orders the LDS traffic for the lane exchange
        if (half == 0) {
#pragma unroll 1
            for (int c2 = 0; c2 < 16; ++c2) {
                const float val = tileD[m * 17 + c2];
                const int   idx = cb + c2;
                if (idx != myrow && val < bd[K_ - 1]) {
                    bd[K_ - 1] = val; bi[K_ - 1] = idx;
#pragma unroll
                    for (int q = K_ - 1; q > 0; --q) {
                        if (bd[q] < bd[q - 1]) {
                            float td = bd[q]; bd[q] = bd[q - 1]; bd[q - 1] = td;
                            int   ti = bi[q]; bi[q] = bi[q - 1]; bi[q - 1] = ti;
                        }
                    }
                }
            }
        }
        __syncthreads();
    }

    if (half == 0) {
        int* o = knn + (size_t)(b * P_ + t * 16 + m) * K_;
#pragma unroll
        for (int q = 0; q < K_; ++q) o[q] = bi[q];
    }
}

// ---------------- Kernel 3: gather neighbor features + max-pool ------------
// One block per sampled point; 256 threads = 256 channels. 16 x 1KB coalesced
// row reads per block, nearly all hitting L2 (feature map is 32MB << 192MB).
__global__ __launch_bounds__(256) void pool_kernel(const float* __restrict__ feat,
                                                   const int* __restrict__ knn,
                                                   float* __restrict__ out_f)
{
    __shared__ int nb[K_];
    const int blk = blockIdx.x;        // b * P_ + p
    const int b   = blk >> 10;
    const int tid = threadIdx.x;

    if (tid < K_) nb[tid] = knn[(size_t)blk * K_ + tid];
    __syncthreads();

    const float* fb = feat + (size_t)b * N_ * C_;
    float m = -3.4e38f;
#pragma unroll
    for (int q = 0; q < K_; ++q)
        m = fmaxf(m, fb[(size_t)nb[q] * C_ + tid]);

    out_f[(size_t)blk * C_ + tid] = m;
}

extern "C" void kernel_launch(void* const* d_in, const int* in_sizes, int n_in,
                              void* d_out, int out_size, void* d_ws, size_t ws_size,
                              hipStream_t stream)
{
    const float* verts = (const float*)d_in[0];   // [8, 4096, 3]
    const float* feat  = (const float*)d_in[1];   // [8, 4096, 256]
    // d_in[2] = pooling_to (1024), d_in[3] = neighbor_num (16): fixed here.

    float* out    = (float*)d_out;                // [8*1024*3] ++ [8*1024*256]
    int*   samp   = (int*)d_ws;                   // 8*1024 ints
    int*   knn    = samp + B_ * P_;               // 8*1024*16 ints
    float* out_f  = out + B_ * P_ * 3;

    fps_kernel <<<B_,            256, 0, stream>>>(verts, out, samp);
    knn_kernel <<<B_ * (P_/16),  32,  0, stream>>>(verts, samp, knn);
    pool_kernel<<<B_ * P_,       256, 0, stream>>>(feat, knn, out_f);
}